// PCKNO_61272003445398
// MI455X (gfx1250) — compile-verified
//
#include <hip/hip_runtime.h>
#include <math.h>

// ---------------- problem constants (from reference setup_inputs) ----------
#define B_   4
#define N_   20000
#define E_   100000
#define C_   128
#define K_   32          // spectral modes
#define KTOT 576         // 128 (h) + 384 (g) + 32 (cos) + 32 (sin)
#define ASTR 36          // LDS A-tile row stride (mult. of 4 -> 16B aligned rows)

typedef __attribute__((ext_vector_type(2))) float v2f;
typedef __attribute__((ext_vector_type(4))) float v4f;
typedef __attribute__((ext_vector_type(8))) float v8f;

__device__ __forceinline__ float gelu_exact(float x) {
    return 0.5f * x * (1.0f + erff(x * 0.70710678118654752f));
}

// ---------------------------------------------------------------------------
// misc small kernels
// ---------------------------------------------------------------------------
__global__ void k_zero_u32(unsigned int* p, int n) {
    int i = blockIdx.x * 256 + threadIdx.x;
    if (i < n) p[i] = 0u;
}

// basis: t = nodes . (modes * invL); bc=cos, bs=sin; wbc/wbs weighted versions
__global__ void k_basis(const float* __restrict__ nodes, const float* __restrict__ nw,
                        const float* __restrict__ modes, const float* __restrict__ latent,
                        float* __restrict__ BC, float* __restrict__ BS,
                        float* __restrict__ WBC, float* __restrict__ WBS) {
    int idx = blockIdx.x * 256 + threadIdx.x;          // over B*N*32, k fastest
    if (idx >= B_ * N_ * K_) return;
    int k  = idx & 31;
    int bn = idx >> 5;
    float t = 0.f;
#pragma unroll
    for (int d = 0; d < 3; ++d) {
        float il = 0.5f + 1.5f / (1.f + expf(-latent[d]));   // INV_L in [0.5,2.0]
        t += nodes[bn * 3 + d] * modes[k * 3 + d] * il;
    }
    float cv = cosf(t), sv = sinf(t);
    float w = nw[bn];
    BC[idx] = cv;  BS[idx] = sv;
    WBC[idx] = cv * w;  WBS[idx] = sv * w;
}

// fc0: h[b,n,c] = x[b,n,:3] . fc0_w[c,:3] + fc0_b[c]
__global__ void k_fc0(const float* __restrict__ x, const float* __restrict__ w,
                      const float* __restrict__ bias, float* __restrict__ H) {
    int idx = blockIdx.x * 256 + threadIdx.x;          // over B*N*C, c fastest
    if (idx >= B_ * N_ * C_) return;
    int c = idx & 127;
    int bn = idx >> 7;
    float a = bias[c];
    a += x[bn * 3 + 0] * w[c * 3 + 0] + x[bn * 3 + 1] * w[c * 3 + 1] + x[bn * 3 + 2] * w[c * 3 + 2];
    H[idx] = a;
}

// transpose fc1_w (f,k) -> FT (k,f)
__global__ void k_fc1t(const float* __restrict__ w, float* __restrict__ FT) {
    int idx = blockIdx.x * 256 + threadIdx.x;
    if (idx >= C_ * C_) return;
    int f = idx & 127, k = idx >> 7;
    FT[k * C_ + f] = w[f * C_ + k];
}

// ---------------------------------------------------------------------------
// CSR build (edges grouped by target node) -- deterministic, rebuilt each call
// ---------------------------------------------------------------------------
__global__ void k_count(const int* __restrict__ edges, int* __restrict__ cnt) {
    int idx = blockIdx.x * 256 + threadIdx.x;          // over B*E
    if (idx >= B_ * E_) return;
    int b = idx / E_;
    int tgt = edges[(size_t)idx * 2 + 0];
    atomicAdd(&cnt[b * N_ + tgt], 1);
}

__global__ void k_scan1(const int* __restrict__ cnt, int* __restrict__ off,
                        int* __restrict__ part, int n) {
    __shared__ int s[256];
    int tid = threadIdx.x, gid = blockIdx.x * 256 + tid;
    int v = (gid < n) ? cnt[gid] : 0;
    s[tid] = v; __syncthreads();
    for (int o = 1; o < 256; o <<= 1) {
        int t = 0; if (tid >= o) t = s[tid - o];
        __syncthreads(); s[tid] += t; __syncthreads();
    }
    if (gid < n) off[gid] = s[tid] - v;                 // exclusive
    if (tid == 255) part[blockIdx.x] = s[255];          // block total
}

__global__ void k_scan2(int* part, int np) {            // np <= 512
    __shared__ int s[512];
    int tid = threadIdx.x;
    int v = (tid < np) ? part[tid] : 0;
    s[tid] = v; __syncthreads();
    for (int o = 1; o < 512; o <<= 1) {
        int t = 0; if (tid >= o) t = s[tid - o];
        __syncthreads(); s[tid] += t; __syncthreads();
    }
    if (tid < np) part[tid] = s[tid] - v;               // exclusive
}

__global__ void k_scan3(int* __restrict__ off, const int* __restrict__ part, int n) {
    int gid = blockIdx.x * 256 + threadIdx.x;
    if (gid < n) off[gid] += part[blockIdx.x];
}

__global__ void k_copy_i32(const int* __restrict__ a, int* __restrict__ b, int n) {
    int i = blockIdx.x * 256 + threadIdx.x;
    if (i < n) b[i] = a[i];
}

__global__ void k_fill(const int* __restrict__ edges, int* __restrict__ cur,
                       int* __restrict__ es) {
    int idx = blockIdx.x * 256 + threadIdx.x;
    if (idx >= B_ * E_) return;
    int b = idx / E_, e = idx % E_;
    int tgt = edges[(size_t)idx * 2 + 0];
    int pos = atomicAdd(&cur[b * N_ + tgt], 1);
    es[pos] = e;
}

// ---------------------------------------------------------------------------
// spectral forward: xc[b,c,k] = sum_n h[b,n,c]*wbc[b,n,k] (and xs, x0)
// grid (chunks, 33, B), 128 threads = c ; partial sums via f32 atomics (~0.8M)
// ---------------------------------------------------------------------------
#define SF_CHUNKS 25
#define SF_LEN    800
__global__ void k_spec_fwd(const float* __restrict__ H, const float* __restrict__ WBC,
                           const float* __restrict__ WBS, const float* __restrict__ nw,
                           float* __restrict__ XC, float* __restrict__ XS,
                           float* __restrict__ X0) {
    int c = threadIdx.x;
    int n0 = blockIdx.x * SF_LEN;
    int k  = blockIdx.y;           // 0..31 = modes, 32 = x0
    int b  = blockIdx.z;
    const float* hb = H + ((size_t)(b * N_) << 7);
    if (k < 32) {
        float ac = 0.f, as = 0.f;
        const float* wc = WBC + (size_t)(b * N_) * 32 + k;
        const float* ws = WBS + (size_t)(b * N_) * 32 + k;
        for (int n = n0; n < n0 + SF_LEN; ++n) {
            float h = hb[((size_t)n << 7) + c];
            ac += h * wc[(size_t)n * 32];
            as += h * ws[(size_t)n * 32];
        }
        atomicAdd(&XC[(b * C_ + c) * 32 + k], ac);
        atomicAdd(&XS[(b * C_ + c) * 32 + k], as);
    } else {
        float a0 = 0.f;
        const float* wv = nw + b * N_;
        for (int n = n0; n < n0 + SF_LEN; ++n)
            a0 += hb[((size_t)n << 7) + c] * wv[n];
        atomicAdd(&X0[b * C_ + c], a0);
    }
}

// fill BMAT rows 0..511: ww^T then gw^T (per batch copy so GEMM indexing is uniform)
__global__ void k_bmat(const float* __restrict__ ww, const float* __restrict__ gww,
                       float* __restrict__ BMAT, int layer) {
    int idx = blockIdx.x * 256 + threadIdx.x;          // B*512*128
    if (idx >= B_ * 512 * C_) return;
    int o = idx & 127;
    int k = (idx >> 7) & 511;
    int b = idx >> 16;
    float v;
    if (k < 128) v = ww[(size_t)layer * C_ * C_ + o * C_ + k];
    else         v = gww[(size_t)layer * C_ * 3 * C_ + o * (3 * C_) + (k - 128)];
    BMAT[((size_t)b * KTOT + k) * C_ + o] = v;
}

// spectral mix: fc = xc*wc - xs*ws ; fs = xs*wc + xc*ws ; f0 = x0*w0
// packs 2*fc / -2*fs into BMAT rows 512..575 and f0+wb+gb into BIAS
__global__ void k_mix(const float* __restrict__ XC, const float* __restrict__ XS,
                      const float* __restrict__ X0,
                      const float* __restrict__ swc, const float* __restrict__ sws,
                      const float* __restrict__ sw0,
                      const float* __restrict__ wb, const float* __restrict__ gb,
                      float* __restrict__ BMAT, float* __restrict__ BIAS, int layer) {
    int idx = blockIdx.x * 256 + threadIdx.x;          // B*128*33, k fastest
    if (idx >= B_ * C_ * 33) return;
    int k = idx % 33;
    int o = (idx / 33) & 127;
    int b = idx / (33 * 128);
    if (k < 32) {
        const float* wc = swc + (size_t)layer * C_ * C_ * 32;
        const float* ws = sws + (size_t)layer * C_ * C_ * 32;
        float fc = 0.f, fs = 0.f;
        for (int c = 0; c < C_; ++c) {
            float xc = XC[(b * C_ + c) * 32 + k];
            float xs = XS[(b * C_ + c) * 32 + k];
            float a = wc[((size_t)c * C_ + o) * 32 + k];
            float s = ws[((size_t)c * C_ + o) * 32 + k];
            fc += xc * a - xs * s;
            fs += xs * a + xc * s;
        }
        BMAT[((size_t)b * KTOT + 512 + k) * C_ + o] =  2.f * fc;
        BMAT[((size_t)b * KTOT + 544 + k) * C_ + o] = -2.f * fs;
    } else {
        const float* w0 = sw0 + (size_t)layer * C_ * C_;
        float f0 = 0.f;
        for (int c = 0; c < C_; ++c) f0 += X0[b * C_ + c] * w0[(size_t)c * C_ + o];
        BIAS[b * C_ + o] = f0 + wb[layer * C_ + o] + gb[layer * C_ + o];
    }
}

// gradient: g[b,n,c*3+d] = sum_{edges with target n} egw[e,d]*(h[src,c]-h[n,c])
// CSR -> no atomics; grid (N,B), 128 threads = c
__global__ void k_grad(const float* __restrict__ H, const int* __restrict__ edges,
                       const float* __restrict__ egw, const int* __restrict__ off,
                       const int* __restrict__ cnt, const int* __restrict__ es,
                       float* __restrict__ G) {
    int c = threadIdx.x;
    int n = blockIdx.x;
    int b = blockIdx.y;
    int row = b * N_ + n;
    int st = off[row], num = cnt[row];
    float ht = H[((size_t)row << 7) + c];
    float a0 = 0.f, a1 = 0.f, a2 = 0.f;
    for (int j = 0; j < num; ++j) {
        int e = es[st + j];
        size_t eb = (size_t)b * E_ + e;
        int s = edges[eb * 2 + 1];
        float d = H[((size_t)(b * N_ + s) << 7) + c] - ht;
        a0 += egw[eb * 3 + 0] * d;
        a1 += egw[eb * 3 + 1] * d;
        a2 += egw[eb * 3 + 2] * d;
    }
    float* g = G + (size_t)row * 384 + c * 3;
    g[0] = a0; g[1] = a1; g[2] = a2;
}

// ---------------------------------------------------------------------------
// fused layer GEMM: Hout[n,o] = sum_{k<576} A[n,k]*BMAT[b,k,o] + BIAS[b,o]
// A[n,:] = [ h[n,0:128] | g[n,0:384] | bc[n,0:32] | bs[n,0:32] ]
// f32 WMMA 16x16x4; block = 8 waves = 32 nodes x 128 outputs.
// A staged via one b128/thread; B staged via four b128/thread into a
// K-pair-interleaved layout: (k,c) -> Bs[(k>>1)*256 + c*2 + (k&1)] so each
// WMMA B operand is one aligned float2 (single ds_load_2addr_b64 per pair).
// ---------------------------------------------------------------------------
__global__ __launch_bounds__(256) void k_gemm(
    const float* __restrict__ H, const float* __restrict__ G,
    const float* __restrict__ BC, const float* __restrict__ BS,
    const float* __restrict__ BMAT, const float* __restrict__ BIAS,
    float* __restrict__ Hout, int do_gelu) {
    __shared__ __align__(16) float As[32 * ASTR];
    __shared__ __align__(16) float Bs[32 * 128];
    int tid = threadIdx.x;
    int b = blockIdx.y;
    int n0 = blockIdx.x * 32;
    int wid = tid >> 5, lane = tid & 31;
    int mi = wid & 1, nj = wid >> 1;
    int m = lane & 15;
    int kb = (lane >> 4) << 1;                 // 0 for lanes 0-15, 2 for 16-31
    int col0 = nj * 32 + m;
    int arow = tid >> 3, aq = (tid & 7) << 2;  // A staging: 32 rows x 8 quads
    v8f acc0 = {}; v8f acc1 = {};
    for (int kc = 0; kc < 18; ++kc) {
        int kg = kc * 32;
        // stage A chunk (32 nodes x 32 K): one float4 per thread, coalesced
        {
            size_t node = (size_t)(b * N_) + n0 + arow;
            v4f v;
            if (kg < 128)      v = *(const v4f*)&H[(node << 7) + kg + aq];
            else if (kg < 512) v = *(const v4f*)&G[node * 384 + (kg - 128) + aq];
            else if (kc == 16) v = *(const v4f*)&BC[(node << 5) + aq];
            else               v = *(const v4f*)&BS[(node << 5) + aq];
            *(v4f*)&As[arow * ASTR + aq] = v;
        }
        // stage B chunk (32 K x 128 out): four float4 per thread, interleaved
#pragma unroll
        for (int r = 0; r < 4; ++r) {
            int lin = r * 256 + tid;           // 1024 float4s
            int row = lin >> 5, c4 = (lin & 31) << 2;
            v4f v = *(const v4f*)&BMAT[((size_t)b * KTOT + kg + row) * C_ + c4];
            int base = (row >> 1) * 256 + c4 * 2 + (row & 1);
            Bs[base + 0] = v.x; Bs[base + 2] = v.y;
            Bs[base + 4] = v.z; Bs[base + 6] = v.w;
        }
        __syncthreads();
#pragma unroll
        for (int ks = 0; ks < 8; ++ks) {
            int kk = ks * 4 + kb;                         // even
            v2f a = *(const v2f*)&As[(mi * 16 + m) * ASTR + kk];
            const float* bp = &Bs[(kk >> 1) * 256 + col0 * 2];
            v2f b0 = *(const v2f*)bp;                     // (B[kk][col0], B[kk+1][col0])
            v2f b1 = *(const v2f*)(bp + 32);              // col1 = col0 + 16
            acc0 = __builtin_amdgcn_wmma_f32_16x16x4_f32(false, a, false, b0, (short)0, acc0, false, false);
            acc1 = __builtin_amdgcn_wmma_f32_16x16x4_f32(false, a, false, b1, (short)0, acc1, false, false);
        }
        __syncthreads();
    }
    int col1 = col0 + 16;
    float bias0 = BIAS[b * C_ + col0];
    float bias1 = BIAS[b * C_ + col1];
    int rbase = mi * 16 + ((lane >> 4) << 3);  // D: vgpr v -> row v (lanes<16) / v+8
#pragma unroll
    for (int v = 0; v < 8; ++v) {
        size_t node = (size_t)(b * N_) + n0 + rbase + v;
        float v0 = acc0[v] + bias0;
        float v1 = acc1[v] + bias1;
        if (do_gelu) { v0 = gelu_exact(v0); v1 = gelu_exact(v1); }
        Hout[(node << 7) + col0] = v0;
        Hout[(node << 7) + col1] = v1;
    }
}

// head: z = gelu(h @ fc1_w^T + fc1_b) ; out = z @ fc2_w^T + fc2_b
__global__ __launch_bounds__(256) void k_head(
    const float* __restrict__ H, const float* __restrict__ FT,
    const float* __restrict__ fc1b, const float* __restrict__ fc2w,
    const float* __restrict__ fc2b, float* __restrict__ out) {
    __shared__ __align__(16) float As[32 * ASTR];
    __shared__ __align__(16) float Bs[32 * 128];
    __shared__ float Zs[32 * 130];
    int tid = threadIdx.x;
    int b = blockIdx.y;
    int n0 = blockIdx.x * 32;
    int wid = tid >> 5, lane = tid & 31;
    int mi = wid & 1, nj = wid >> 1;
    int m = lane & 15;
    int kb = (lane >> 4) << 1;
    int col0 = nj * 32 + m;
    int arow = tid >> 3, aq = (tid & 7) << 2;
    v8f acc0 = {}; v8f acc1 = {};
    for (int kc = 0; kc < 4; ++kc) {
        int kg = kc * 32;
        {
            size_t node = (size_t)(b * N_) + n0 + arow;
            v4f v = *(const v4f*)&H[(node << 7) + kg + aq];
            *(v4f*)&As[arow * ASTR + aq] = v;
        }
#pragma unroll
        for (int r = 0; r < 4; ++r) {
            int lin = r * 256 + tid;
            int row = lin >> 5, c4 = (lin & 31) << 2;
            v4f v = *(const v4f*)&FT[(size_t)(kg + row) * C_ + c4];
            int base = (row >> 1) * 256 + c4 * 2 + (row & 1);
            Bs[base + 0] = v.x; Bs[base + 2] = v.y;
            Bs[base + 4] = v.z; Bs[base + 6] = v.w;
        }
        __syncthreads();
#pragma unroll
        for (int ks = 0; ks < 8; ++ks) {
            int kk = ks * 4 + kb;
            v2f a = *(const v2f*)&As[(mi * 16 + m) * ASTR + kk];
            const float* bp = &Bs[(kk >> 1) * 256 + col0 * 2];
            v2f b0 = *(const v2f*)bp;
            v2f b1 = *(const v2f*)(bp + 32);
            acc0 = __builtin_amdgcn_wmma_f32_16x16x4_f32(false, a, false, b0, (short)0, acc0, false, false);
            acc1 = __builtin_amdgcn_wmma_f32_16x16x4_f32(false, a, false, b1, (short)0, acc1, false, false);
        }
        __syncthreads();
    }
    int col1 = col0 + 16;
    float b0v = fc1b[col0], b1v = fc1b[col1];
    int rbase = mi * 16 + ((lane >> 4) << 3);
#pragma unroll
    for (int v = 0; v < 8; ++v) {
        Zs[(rbase + v) * 130 + col0] = gelu_exact(acc0[v] + b0v);
        Zs[(rbase + v) * 130 + col1] = gelu_exact(acc1[v] + b1v);
    }
    __syncthreads();
    if (tid < 32) {
        float a = fc2b[0];
        for (int f = 0; f < C_; ++f) a += Zs[tid * 130 + f] * fc2w[f];
        out[(size_t)b * N_ + n0 + tid] = a;
    }
}

// ---------------------------------------------------------------------------
extern "C" void kernel_launch(void* const* d_in, const int* in_sizes, int n_in,
                              void* d_out, int out_size, void* d_ws, size_t ws_size,
                              hipStream_t stream) {
    (void)in_sizes; (void)n_in; (void)out_size; (void)ws_size;
    const float* x      = (const float*)d_in[0];
    // d_in[1] = node_mask (unused by reference)
    const float* nodes  = (const float*)d_in[2];
    const float* nw     = (const float*)d_in[3];
    const int*   edges  = (const int*)  d_in[4];
    const float* egw    = (const float*)d_in[5];
    const float* modes  = (const float*)d_in[6];
    const float* latent = (const float*)d_in[7];
    const float* fc0w   = (const float*)d_in[8];
    const float* fc0b   = (const float*)d_in[9];
    const float* swc    = (const float*)d_in[10];
    const float* sws    = (const float*)d_in[11];
    const float* sw0    = (const float*)d_in[12];
    const float* ww     = (const float*)d_in[13];
    const float* wb     = (const float*)d_in[14];
    const float* gw     = (const float*)d_in[15];
    const float* gb     = (const float*)d_in[16];
    const float* fc1w   = (const float*)d_in[17];
    const float* fc1b   = (const float*)d_in[18];
    const float* fc2w   = (const float*)d_in[19];
    const float* fc2b   = (const float*)d_in[20];
    float* out = (float*)d_out;

    // workspace carve-up (~250 MB)
    char* base = (char*)d_ws;
    size_t off = 0;
    auto alloc = [&](size_t bytes) -> char* {
        char* p = base + off;
        off = (off + bytes + 255) & ~(size_t)255;
        return p;
    };
    float* H0   = (float*)alloc((size_t)B_ * N_ * C_ * 4);
    float* H1   = (float*)alloc((size_t)B_ * N_ * C_ * 4);
    float* G    = (float*)alloc((size_t)B_ * N_ * 384 * 4);
    float* BC   = (float*)alloc((size_t)B_ * N_ * 32 * 4);
    float* BS   = (float*)alloc((size_t)B_ * N_ * 32 * 4);
    float* WBC  = (float*)alloc((size_t)B_ * N_ * 32 * 4);
    float* WBS  = (float*)alloc((size_t)B_ * N_ * 32 * 4);
    float* XC   = (float*)alloc((size_t)(2 * B_ * C_ * 32 + B_ * C_) * 4); // XC|XS|X0
    float* XS   = XC + B_ * C_ * 32;
    float* X0   = XS + B_ * C_ * 32;
    float* BMAT = (float*)alloc((size_t)B_ * KTOT * C_ * 4);
    float* BIAS = (float*)alloc((size_t)B_ * C_ * 4);
    float* FT   = (float*)alloc((size_t)C_ * C_ * 4);
    int*   CNT  = (int*)alloc((size_t)B_ * N_ * 4);
    int*   OFF  = (int*)alloc((size_t)B_ * N_ * 4);
    int*   CUR  = (int*)alloc((size_t)B_ * N_ * 4);
    int*   ES   = (int*)alloc((size_t)B_ * E_ * 4);
    int*   PART = (int*)alloc(512 * 4);

    // precompute basis, fc0, fc1^T
    k_basis<<<(B_ * N_ * K_ + 255) / 256, 256, 0, stream>>>(nodes, nw, modes, latent, BC, BS, WBC, WBS);
    k_fc0<<<(B_ * N_ * C_ + 255) / 256, 256, 0, stream>>>(x, fc0w, fc0b, H0);
    k_fc1t<<<(C_ * C_ + 255) / 256, 256, 0, stream>>>(fc1w, FT);

    // CSR by target (rebuilt every call; deterministic)
    int nrows = B_ * N_;
    int nblk = (nrows + 255) / 256;   // 313
    k_zero_u32<<<nblk, 256, 0, stream>>>((unsigned int*)CNT, nrows);
    k_count<<<(B_ * E_ + 255) / 256, 256, 0, stream>>>(edges, CNT);
    k_scan1<<<nblk, 256, 0, stream>>>(CNT, OFF, PART, nrows);
    k_scan2<<<1, 512, 0, stream>>>(PART, nblk);
    k_scan3<<<nblk, 256, 0, stream>>>(OFF, PART, nrows);
    k_copy_i32<<<nblk, 256, 0, stream>>>(OFF, CUR, nrows);
    k_fill<<<(B_ * E_ + 255) / 256, 256, 0, stream>>>(edges, CUR, ES);

    float* hcur = H0;
    float* hnext = H1;
    for (int layer = 0; layer < 3; ++layer) {
        int nxf = 2 * B_ * C_ * 32 + B_ * C_;
        k_zero_u32<<<(nxf + 255) / 256, 256, 0, stream>>>((unsigned int*)XC, nxf);
        dim3 gsf(SF_CHUNKS, 33, B_);
        k_spec_fwd<<<gsf, 128, 0, stream>>>(hcur, WBC, WBS, nw, XC, XS, X0);
        k_bmat<<<(B_ * 512 * C_ + 255) / 256, 256, 0, stream>>>(ww, gw, BMAT, layer);
        k_mix<<<(B_ * C_ * 33 + 255) / 256, 256, 0, stream>>>(XC, XS, X0, swc, sws, sw0,
                                                              wb, gb, BMAT, BIAS, layer);
        dim3 gg(N_, B_);
        k_grad<<<gg, 128, 0, stream>>>(hcur, edges, egw, OFF, CNT, ES, G);
        dim3 ge(N_ / 32, B_);
        k_gemm<<<ge, 256, 0, stream>>>(hcur, G, BC, BS, BMAT, BIAS, hnext, (layer < 2) ? 1 : 0);
        float* t = hcur; hcur = hnext; hnext = t;
    }
    dim3 gh(N_ / 32, B_);
    k_head<<<gh, 256, 0, stream>>>(hcur, FT, fc1b, fc2w, fc2b, out);
}